// mem_update_MSF_66477503807493
// MI455X (gfx1250) — compile-verified
//
#include <hip/hip_runtime.h>
#include <cstdint>

// LIF multi-synaptic-fire temporal scan.
// x:   [T=8, N] float32, N = B*C*H*W = 4,194,304
// out: [T=8, N] float32 spike counts in {0,1,2,3,4}
//
// HBM-bound streaming kernel: ~268 MB total traffic -> ~11.5us at 23.3 TB/s.
// CDNA5 async global->LDS DMA (ASYNCcnt), pipelined per-plane waits
// (async loads complete in order), register recurrence, NT b128 stores.

typedef float v4f __attribute__((ext_vector_type(4)));

#define T_STEPS 8
#define BLOCK   256

__global__ __launch_bounds__(BLOCK) void lif_msf_scan(
    const float* __restrict__ x,
    float* __restrict__ out,
    int N /* elements per time plane */) {

    // 8 planes x 256 lanes x 16B = 32 KB LDS per workgroup
    __shared__ v4f tile[T_STEPS][BLOCK];

    const int tid = threadIdx.x;
    const long long i0 = ((long long)blockIdx.x * BLOCK + tid) * 4;
    if (i0 >= N) return;

    // LDS byte offset of this lane's plane-0 slot (low 32 bits of flat ptr).
    uint32_t lds0 = (uint32_t)(uintptr_t)&tile[0][tid];

    // ---- Phase 1: issue all 8 async DMA loads back-to-back (8-deep MLP) ----
#pragma unroll
    for (int t = 0; t < T_STEPS; ++t) {
        unsigned long long gaddr =
            (unsigned long long)(uintptr_t)(x + (long long)t * N + i0);
        uint32_t laddr = lds0 + (uint32_t)(t * BLOCK * sizeof(v4f));
        // One-touch input stream: NT so L2 doesn't retain it.
        asm volatile("global_load_async_to_lds_b128 %0, %1, off th:TH_LOAD_NT"
                     :
                     : "v"(laddr), "v"(gaddr)
                     : "memory");
    }

    // ---- Phase 2: recurrence over T; wait only for the plane we consume ----
    // Async loads from this wave return done IN ORDER, so ASYNCcnt <= 7-t
    // guarantees planes 0..t have landed. Compute/stores of early planes
    // overlap the tail DMAs.
    float m0 = 0.f, m1 = 0.f, m2 = 0.f, m3 = 0.f;        // membrane
    bool  f0 = false, f1 = false, f2 = false, f3 = false; // fired last step

#pragma unroll
    for (int t = 0; t < T_STEPS; ++t) {
        asm volatile("s_wait_asynccnt %0" :: "n"(T_STEPS - 1 - t) : "memory");

        v4f v = tile[t][tid]; // ds_load_b128

        // mem = fired ? x_t : 0.25*mem + x_t  (== mem_old*DECAY*(1-mask)+x_t)
        m0 = f0 ? v.x : __builtin_fmaf(m0, 0.25f, v.x);
        m1 = f1 ? v.y : __builtin_fmaf(m1, 0.25f, v.y);
        m2 = f2 ? v.z : __builtin_fmaf(m2, 0.25f, v.z);
        m3 = f3 ? v.w : __builtin_fmaf(m3, 0.25f, v.w);

        // spike count: sum of Heaviside at thresholds 0.5 + d, d=0..3
        v4f s;
        s.x = (m0 >= 0.5f ? 1.f : 0.f) + (m0 >= 1.5f ? 1.f : 0.f) +
              (m0 >= 2.5f ? 1.f : 0.f) + (m0 >= 3.5f ? 1.f : 0.f);
        s.y = (m1 >= 0.5f ? 1.f : 0.f) + (m1 >= 1.5f ? 1.f : 0.f) +
              (m1 >= 2.5f ? 1.f : 0.f) + (m1 >= 3.5f ? 1.f : 0.f);
        s.z = (m2 >= 0.5f ? 1.f : 0.f) + (m2 >= 1.5f ? 1.f : 0.f) +
              (m2 >= 2.5f ? 1.f : 0.f) + (m2 >= 3.5f ? 1.f : 0.f);
        s.w = (m3 >= 0.5f ? 1.f : 0.f) + (m3 >= 1.5f ? 1.f : 0.f) +
              (m3 >= 2.5f ? 1.f : 0.f) + (m3 >= 3.5f ? 1.f : 0.f);

        f0 = (m0 >= 0.5f);
        f1 = (m1 >= 0.5f);
        f2 = (m2 >= 0.5f);
        f3 = (m3 >= 0.5f);

        // one-touch output: NT store (working set > 192 MB L2)
        __builtin_nontemporal_store(s, (v4f*)(out + (long long)t * N + i0));
    }
}

extern "C" void kernel_launch(void* const* d_in, const int* in_sizes, int n_in,
                              void* d_out, int out_size, void* d_ws, size_t ws_size,
                              hipStream_t stream) {
    (void)n_in; (void)out_size; (void)d_ws; (void)ws_size;

    const float* x = (const float*)d_in[0];
    float* out = (float*)d_out;

    const int total = in_sizes[0];        // T * N
    const int N = total / T_STEPS;        // elements per time plane (4,194,304)

    const int threads = BLOCK;
    const int vec_elems = N / 4;          // one float4 per thread
    const int blocks = (vec_elems + threads - 1) / threads; // 4096

    lif_msf_scan<<<blocks, threads, 0, stream>>>(x, out, N);
}